// Decode_91207925498244
// MI455X (gfx1250) — compile-verified
//
#include <hip/hip_runtime.h>
#include <cstdint>

#define NBATCH   16
#define N4       10816          // 104*104
#define N16      676            // 26*26
#define NTOT     (N4 + N16)
#define TOPK     2048
#define NTHREADS 256
#define CONF_THR 0.8f
#define TILE_FLT 1024           // floats per async tile (4 KB)

// Truncate a generic pointer to a __shared__ object to its 32-bit LDS byte
// offset (generic LDS addresses are {SHARED_BASE, offset32}; low 32 bits are
// the LDS address per the flat-aperture rules).
__device__ __forceinline__ unsigned lds_off(const void* p) {
  return (unsigned)(unsigned long long)p;
}

// CDNA5 async DMA: 16B per lane, global -> LDS, tracked by ASYNCcnt.
__device__ __forceinline__ void async_chunk_b128(unsigned lds, const float* g) {
  asm volatile("global_load_async_to_lds_b128 %0, %1, off"
               :: "v"(lds), "v"((unsigned long long)(uintptr_t)g)
               : "memory");
}

__global__ __launch_bounds__(NTHREADS)
void yolo_nms_fused(const float* __restrict__ feat4,
                    const float* __restrict__ feat16,
                    float* __restrict__ out) {
  // key region doubles as (x1,y1) pairs after the sort
  __shared__ __align__(16) unsigned long long keyA[TOPK];     // 16 KB
  __shared__ __align__(16) float sX2[TOPK];                   //  8 KB
  __shared__ __align__(16) float sY2[TOPK];                   //  8 KB
  __shared__ __align__(16) float sP[TOPK];                    //  8 KB
  __shared__ __align__(16) float tileBuf[2][TILE_FLT];        //  8 KB
  __shared__ __align__(16) float dumpSlot[4];                 // DMA spill for clamped lanes
  __shared__ unsigned char sSup[TOPK];                        //  2 KB
  __shared__ int sCnt;
  __shared__ int sNext;

  const int tid = threadIdx.x;
  const int b   = blockIdx.x;

  if (tid == 0) sCnt = 0;
  __syncthreads();

  const float* conf4  = feat4  + ((size_t)b * 5 + 4) * N4;
  const float* conf16 = feat16 + ((size_t)b * 5 + 4) * N16;

  // 11 tiles cover feat4's conf channel (10 full + 576-float tail), 1 tile covers feat16's.
  const int numTiles = 12;
  auto tileSrc = [&](int t, const float*& src, int& nflt, int& nbase) {
    if (t < 11) { src = conf4 + t * TILE_FLT;
                  nflt = (t == 10) ? (N4 - 10 * TILE_FLT) : TILE_FLT;
                  nbase = t * TILE_FLT; }
    else        { src = conf16; nflt = N16; nbase = N4; }
  };
  auto issueTile = [&](int t) {
    const float* src; int nflt, nbase;
    tileSrc(t, src, nflt, nbase);
    int nChunks = nflt >> 2;                        // 16B chunks (all counts %4==0)
    int cc = (tid < nChunks) ? tid : (nChunks - 1); // clamp -> EXEC stays full, ASYNCcnt deterministic
    unsigned lds = (tid < nChunks)
        ? (lds_off(&tileBuf[t & 1][0]) + (unsigned)tid * 16u)
        : lds_off(&dumpSlot[0]);
    async_chunk_b128(lds, src + (size_t)cc * 4);
  };

  // ---- Phase 1: double-buffered async scan of the confidence channel ----
  issueTile(0);
  for (int t = 0; t < numTiles; ++t) {
    if (t + 1 < numTiles) {
      issueTile(t + 1);                                   // overlap next DMA with this scan
      asm volatile("s_wait_asynccnt 1" ::: "memory");     // tile t complete (in-order)
    } else {
      asm volatile("s_wait_asynccnt 0" ::: "memory");
    }
    __syncthreads();
    const float* src; int nflt, nbase;
    tileSrc(t, src, nflt, nbase);
    const float* buf = tileBuf[t & 1];
    for (int k = tid; k < nflt; k += NTHREADS) {
      float p = 1.0f / (1.0f + expf(-buf[k]));
      if (p > CONF_THR) {
        unsigned n = (unsigned)(nbase + k);
        // sortable key: (score desc, index asc) under a descending u64 sort
        unsigned long long key =
            ((unsigned long long)(__float_as_uint(p) | 0x80000000u) << 32) |
            (unsigned long long)(~n);
        int slot = atomicAdd(&sCnt, 1);
        if (slot < TOPK) keyA[slot] = key;
      }
    }
    __syncthreads();                                      // buffer t&1 free for reuse
  }

  int cnt = sCnt; if (cnt > TOPK) cnt = TOPK;
  for (int t = cnt + tid; t < TOPK; t += NTHREADS) keyA[t] = 0ull;

  // ---- Phase 2: bitonic sort, descending (reproduces stable top_k) ----
  for (unsigned k = 2; k <= TOPK; k <<= 1) {
    for (unsigned j = k >> 1; j > 0; j >>= 1) {
      __syncthreads();
      for (unsigned i = tid; i < TOPK; i += NTHREADS) {
        unsigned l = i ^ j;
        if (l > i) {
          unsigned long long a = keyA[i], c2 = keyA[l];
          bool descBlock = ((i & k) == 0);
          if (descBlock ? (a < c2) : (a > c2)) { keyA[i] = c2; keyA[l] = a; }
        }
      }
    }
  }
  __syncthreads();

  // ---- Phase 3: decode selected boxes into LDS corner arrays ----
  float* sXY = (float*)keyA;   // alias: slot t -> (x1,y1); safe, same-thread read-then-write
  for (int t = tid; t < TOPK; t += NTHREADS) {
    unsigned long long key = keyA[t];
    float X1 = 0.f, Y1 = 0.f, X2 = 0.f, Y2 = 0.f, P = 0.f;
    if (key != 0ull) {
      unsigned n = ~(unsigned)(key & 0xFFFFFFFFull);
      P = __uint_as_float((unsigned)(key >> 32) ^ 0x80000000u);
      float f0, f1, f2, f3, scl, aw, ah; int row, col;
      if (n < N4) {
        const float* base = feat4 + (size_t)b * 5 * N4;
        f0 = base[n]; f1 = base[N4 + n]; f2 = base[2 * N4 + n]; f3 = base[3 * N4 + n];
        row = (int)(n / 104u); col = (int)(n % 104u);
        scl = 4.0f;  aw = 1.25f;  ah = 1.25f;
      } else {
        unsigned m = n - N4;
        const float* base = feat16 + (size_t)b * 5 * N16;
        f0 = base[m]; f1 = base[N16 + m]; f2 = base[2 * N16 + m]; f3 = base[3 * N16 + m];
        row = (int)(m / 26u); col = (int)(m % 26u);
        scl = 16.0f; aw = 2.125f; ah = 1.625f;
      }
      float sx = 1.0f / (1.0f + expf(-f0));
      float sy = 1.0f / (1.0f + expf(-f1));
      float bw = expf(f2) * aw * scl;               // (exp*anchor)*scale, as reference
      float bh = expf(f3) * ah * scl;
      float bx = ((float)row + sx) * scl;           // gx indexes rows (reference quirk)
      float by = ((float)col + sy) * scl;
      X1 = bx - 0.5f * bw; Y1 = by - 0.5f * bh;
      X2 = bx + 0.5f * bw; Y2 = by + 0.5f * bh;
      sSup[t] = 0;
    } else {
      sSup[t] = 1;                                  // filler: pre-suppressed, zero output
    }
    sXY[2 * t] = X1; sXY[2 * t + 1] = Y1; sX2[t] = X2; sY2[t] = Y2; sP[t] = P;
  }
  __syncthreads();

  // ---- Phase 4: greedy NMS, survivor-skipping (iou>0 <=> iw>0 && ih>0) ----
  int i = -1;
  while (true) {
    if (tid == 0) sNext = TOPK;
    __syncthreads();
    int local = TOPK;
    for (int j = i + 1 + tid; j < TOPK; j += NTHREADS)
      if (!sSup[j]) { local = j; break; }
    if (local < TOPK) atomicMin(&sNext, local);
    __syncthreads();
    i = sNext;
    if (i >= TOPK) break;
    float X1 = sXY[2 * i], Y1 = sXY[2 * i + 1], X2 = sX2[i], Y2 = sY2[i];
    for (int j = i + 1 + tid; j < TOPK; j += NTHREADS) {
      if (!sSup[j]) {
        float iw = fminf(X2, sX2[j]) - fmaxf(X1, sXY[2 * j]);
        float ih = fminf(Y2, sY2[j]) - fmaxf(Y1, sXY[2 * j + 1]);
        if (iw > 0.0f && ih > 0.0f) sSup[j] = 1;
      }
    }
    __syncthreads();
  }

  // ---- Phase 5: write (boxes*keep, keep) ----
  float* outKeep = out + (size_t)NBATCH * TOPK * 5;
  for (int t = tid; t < TOPK; t += NTHREADS) {
    bool kp = (sP[t] > CONF_THR) && (sSup[t] == 0);
    float m = kp ? 1.0f : 0.0f;
    size_t o = ((size_t)b * TOPK + t) * 5;
    out[o + 0] = m * sXY[2 * t];
    out[o + 1] = m * sXY[2 * t + 1];
    out[o + 2] = m * sX2[t];
    out[o + 3] = m * sY2[t];
    out[o + 4] = m * sP[t];
    outKeep[(size_t)b * TOPK + t] = m;
  }
}

extern "C" void kernel_launch(void* const* d_in, const int* in_sizes, int n_in,
                              void* d_out, int out_size, void* d_ws, size_t ws_size,
                              hipStream_t stream) {
  (void)in_sizes; (void)n_in; (void)d_ws; (void)ws_size; (void)out_size;
  const float* feat4  = (const float*)d_in[0];
  const float* feat16 = (const float*)d_in[1];
  float* out = (float*)d_out;
  yolo_nms_fused<<<dim3(NBATCH), dim3(NTHREADS), 0, stream>>>(feat4, feat16, out);
}